// EMloss_88802743812528
// MI455X (gfx1250) — compile-verified
//
#include <hip/hip_runtime.h>

typedef __attribute__((ext_vector_type(2))) float v2f;
typedef __attribute__((ext_vector_type(8))) float v8f;

#define EPS_F 1e-8f

constexpr int Bsz = 16, Cch = 19, Hh = 512, Ww = 512;
constexpr int HW   = Hh * Ww;                       // 262144 = 2^18
constexpr int NPIX = Bsz * HW;                      // 4194304
constexpr size_t NPRED = (size_t)Bsz * Cch * HW;    // 79691776
constexpr int THREADS = 256;                        // 8 waves (wave32)
constexpr int NBLOCKS = 2048;
constexpr int TILES   = NPIX / (THREADS * NBLOCKS); // 8 tiles/block, contiguous span

// ---------------------------------------------------------------------------
// Wave32 sum using the matrix pipe:
//   D(16x16) = A(16x4) x B(4x16) + 0, with A holding the 32 lane partials
//   (lane i -> A[i][0], lane 16+i -> A[i][2]; K=1,3 zero) and B = ones.
//   => D[i][j] = s(i) + s(i+16) for every j.
//   Per-lane sum of the 8 D VGPRs gives rows 0-7 summed in lanes 0-15 and
//   rows 8-15 summed in lanes 16-31; one SWAPX16 ds_swizzle folds halves.
// EXEC must be all ones at the call site (WMMA restriction).
// ---------------------------------------------------------------------------
__device__ __forceinline__ float wave_sum32(float s) {
  v2f a; a[0] = s;    a[1] = 0.0f;
  v2f b; b[0] = 1.0f; b[1] = 1.0f;
  v8f c = {};
  v8f d = __builtin_amdgcn_wmma_f32_16x16x4_f32(false, a, false, b,
                                                (short)0, c, false, false);
  float v = ((d[0] + d[1]) + (d[2] + d[3])) + ((d[4] + d[5]) + (d[6] + d[7]));
  // group-of-32 swizzle, xor_mask=0x10 (SWAPX16): fold lanes 0-15 <-> 16-31
  v += __int_as_float(__builtin_amdgcn_ds_swizzle(__float_as_int(v), 0x401F));
  return v;  // full wave sum in every lane
}

// ---------------------------------------------------------------------------
// Main kernel: streaming softmax + EM loss. One thread = one pixel per tile.
// ---------------------------------------------------------------------------
__global__ __launch_bounds__(THREADS) void emloss_main(
    const float* __restrict__ logits,   // [B,C,H,W]
    const int*   __restrict__ targets,  // [B,H,W]
    const float* __restrict__ cmat,     // [C,C]
    float*       __restrict__ out,      // pred [B,C,H,W] (+ scalar slot later)
    float*       __restrict__ partials) // [NBLOCKS]
{
  // arr[t][c] = ncm[c][t] and log(ncm[c][t]+eps): 19x19 tables, padded stride
  __shared__ float ncmT[Cch][20];
  __shared__ float lpT [Cch][20];
  __shared__ float wpart[THREADS / 32];

  const int tid = threadIdx.x;

  // Per-block table build: thread j computes row j of ncm = softmax(cm row j)
  if (tid < Cch) {
    float row[Cch];
    float m = -3.4e38f;
    #pragma unroll
    for (int k = 0; k < Cch; ++k) { row[k] = cmat[tid * Cch + k]; m = fmaxf(m, row[k]); }
    float s = 0.0f;
    #pragma unroll
    for (int k = 0; k < Cch; ++k) { row[k] = __expf(row[k] - m); s += row[k]; }
    const float inv = 1.0f / s;
    #pragma unroll
    for (int k = 0; k < Cch; ++k) {
      const float v = row[k] * inv;          // ncm[tid][k]
      ncmT[k][tid] = v;                      // transposed: ncmT[t][c] = ncm[c][t]
      lpT [k][tid] = __logf(v + EPS_F);      // log(pre_p + eps) table
    }
  }
  __syncthreads();

  const int pbase = blockIdx.x * (TILES * THREADS) + tid;  // contiguous block span
  float loss_acc = 0.0f;

  for (int tile = 0; tile < TILES; ++tile) {
    const int p  = pbase + tile * THREADS;
    const int b  = p >> 18;                  // HW = 2^18
    const int hw = p & (HW - 1);
    const int t  = targets[p];
    const float* lp = logits + (size_t)b * Cch * HW + hw;

    // 19 coalesced streaming loads (no reuse -> non-temporal)
    float x[Cch];
    #pragma unroll
    for (int cc = 0; cc < Cch; ++cc)
      x[cc] = __builtin_nontemporal_load(lp + cc * HW);

    // Prefetch next tile's channel planes: 1 lane per 64B line is enough
    if ((tile + 1 < TILES) && ((tid & 15) == 0)) {
      const float* np = lp + THREADS;
      #pragma unroll
      for (int cc = 0; cc < Cch; ++cc)
        __builtin_prefetch(np + cc * HW, 0, 1);   // -> global_prefetch_b8
    }

    // Stable softmax over C
    float m = x[0];
    #pragma unroll
    for (int cc = 1; cc < Cch; ++cc) m = fmaxf(m, x[cc]);
    float s = 0.0f;
    #pragma unroll
    for (int cc = 0; cc < Cch; ++cc) { x[cc] = __expf(x[cc] - m); s += x[cc]; }
    const float inv = 1.0f / s;

    // pred out + fused q-normalized loss:
    //   w[c] = ncm[c][t]*pred[c];  loss = (sum w*(log(pre_p+e)+log(pred+e))) / sum w
    float* po = out + (size_t)b * Cch * HW + hw;
    float qs = 0.0f, acc = 0.0f;
    #pragma unroll
    for (int cc = 0; cc < Cch; ++cc) {
      const float pr = x[cc] * inv;
      __builtin_nontemporal_store(pr, po + cc * HW);
      const float w = ncmT[t][cc] * pr;
      qs  += w;
      acc += w * (lpT[t][cc] + __logf(pr + EPS_F));
    }
    loss_acc += acc / qs;
  }

  // Deterministic block reduction: WMMA wave sum -> LDS -> thread 0
  const float wv = wave_sum32(loss_acc);
  if ((tid & 31) == 0) wpart[tid >> 5] = wv;
  __syncthreads();
  if (tid == 0) {
    float s = 0.0f;
    #pragma unroll
    for (int i = 0; i < THREADS / 32; ++i) s += wpart[i];
    partials[blockIdx.x] = s;
  }
}

// ---------------------------------------------------------------------------
// Finalize: sum 2048 block partials in a fixed order, write -mean scalar.
// ---------------------------------------------------------------------------
__global__ __launch_bounds__(THREADS) void emloss_finalize(
    const float* __restrict__ partials, float* __restrict__ out)
{
  __shared__ float wpart[THREADS / 32];
  const int tid = threadIdx.x;

  float a = 0.0f;
  #pragma unroll
  for (int k = 0; k < NBLOCKS / THREADS; ++k)
    a += partials[tid + k * THREADS];

  const float wv = wave_sum32(a);
  if ((tid & 31) == 0) wpart[tid >> 5] = wv;
  __syncthreads();
  if (tid == 0) {
    float s = 0.0f;
    #pragma unroll
    for (int i = 0; i < THREADS / 32; ++i) s += wpart[i];
    out[NPRED] = -s * (1.0f / (float)NPIX);   // batch_loss after pred block
  }
}

// ---------------------------------------------------------------------------
extern "C" void kernel_launch(void* const* d_in, const int* in_sizes, int n_in,
                              void* d_out, int out_size, void* d_ws, size_t ws_size,
                              hipStream_t stream) {
  (void)in_sizes; (void)n_in; (void)out_size; (void)ws_size;
  const float* logits  = (const float*)d_in[0];
  const int*   targets = (const int*)d_in[1];
  const float* cmat    = (const float*)d_in[2];
  float* out      = (float*)d_out;
  float* partials = (float*)d_ws;   // 2048 floats of scratch

  emloss_main    <<<dim3(NBLOCKS), dim3(THREADS), 0, stream>>>(logits, targets, cmat, out, partials);
  emloss_finalize<<<dim3(1),       dim3(THREADS), 0, stream>>>(partials, out);
}